// RQLHypergraphEngine_12463995093874
// MI455X (gfx1250) — compile-verified
//
#include <hip/hip_runtime.h>
#include <hip/hip_bf16.h>
#include <math.h>

typedef __attribute__((ext_vector_type(2))) float v2f;
typedef __attribute__((ext_vector_type(8))) float v8f;

#define DIM 1024

// ---------------------------------------------------------------------------
// Kernel 1: u[k] = dot(Wq[k, 0:1024], Ws[0:1024])   (k = 0..2047)
// One wave per row; coalesced float4 loads; shuffle reduction (wave32).
// ---------------------------------------------------------------------------
__global__ void wq_dot_kernel(const float* __restrict__ Wq,
                              const float* __restrict__ Ws,
                              float* __restrict__ u) {
    const int wave = (blockIdx.x * blockDim.x + threadIdx.x) >> 5; // 0..2047
    const int lane = threadIdx.x & 31;
    const float* row = Wq + (size_t)wave * DIM;
    float acc = 0.0f;
#pragma unroll
    for (int i = 0; i < 8; ++i) {
        const int j = i * 128 + lane * 4;
        float4 w = *(const float4*)(row + j);
        float4 s = *(const float4*)(Ws + j);
        acc = fmaf(w.x, s.x, acc);
        acc = fmaf(w.y, s.y, acc);
        acc = fmaf(w.z, s.z, acc);
        acc = fmaf(w.w, s.w, acc);
    }
#pragma unroll
    for (int m = 16; m > 0; m >>= 1) acc += __shfl_xor(acc, m, 32);
    if (lane == 0) u[wave] = acc;
}

// ---------------------------------------------------------------------------
// Kernel 2: q_const = cat(s_emb, r_emb) . u  +  bq . Ws[0:1024]  +  bs
// Single block, LDS tree reduction. (q_term of the reference, refactored:
// (cat@Wq + bq)@wsq == cat@(Wq@wsq) + bq@wsq.)
// ---------------------------------------------------------------------------
__global__ void qconst_kernel(const float* __restrict__ ent,
                              const float* __restrict__ rel,
                              const int* __restrict__ subject,
                              const int* __restrict__ relation,
                              const float* __restrict__ u,
                              const float* __restrict__ Ws,
                              const float* __restrict__ bq,
                              const float* __restrict__ bs,
                              float* __restrict__ qc) {
    __shared__ float red[256];
    const int t = threadIdx.x;
    const float* s_emb = ent + (size_t)subject[0] * DIM;
    const float* r_emb = rel + (size_t)relation[0] * DIM;
    float acc = 0.0f;
    for (int k = t; k < DIM; k += 256) acc = fmaf(s_emb[k], u[k], acc);
    for (int k = t; k < DIM; k += 256) acc = fmaf(r_emb[k], u[DIM + k], acc);
    for (int j = t; j < DIM; j += 256) acc = fmaf(bq[j], Ws[j], acc);
    red[t] = acc;
    __syncthreads();
#pragma unroll
    for (int s = 128; s > 0; s >>= 1) {
        if (t < s) red[t] += red[t + s];
        __syncthreads();
    }
    if (t == 0) qc[0] = red[0] + bs[0];
}

// ---------------------------------------------------------------------------
// Kernel 3: ent_dot[v] = dot(entity_embed[v, :], w_c)  for all vocab rows,
// 16 rows per wave via V_WMMA_F32_16X16X4_F32 accumulation over K.
//   A (16x4 f32, 2 VGPRs):  lane L -> row rowBase+(L&15), K = k0+2*(L>>4)+{0,1}
//   B (4x16  f32, 2 VGPRs): w chunk replicated across columns; lane-half K
//                           split matches A, so the same float2 of w serves it.
//   D (16x16 f32): column N=0 lives in lanes 0 and 16 -> those lanes hold the
//                  16 dot products in their 8 accumulator VGPRs.
// One full wave per block => EXEC all-ones for WMMA; OOB rows clamp on load.
// ---------------------------------------------------------------------------
__global__ void ent_dot_kernel(const float* __restrict__ ent,
                               const float* __restrict__ wc,
                               float* __restrict__ ent_dot,
                               int n_ent) {
    const int lane = threadIdx.x;          // blockDim.x == 32
    const int rowBase = blockIdx.x * 16;
    const int half = lane >> 4;            // 0: K pair {0,1}, 1: K pair {2,3}
    const int m = lane & 15;
    int row = rowBase + m;
    if (row > n_ent - 1) row = n_ent - 1;  // clamp (keep EXEC all-ones)
    const float* rowp = ent + (size_t)row * DIM + 2 * half;
    const float* wp = wc + 2 * half;

    v8f acc = {};
#pragma unroll 8
    for (int k0 = 0; k0 < DIM; k0 += 4) {
        v2f a = *(const v2f*)(rowp + k0);
        v2f b = *(const v2f*)(wp + k0);
        acc = __builtin_amdgcn_wmma_f32_16x16x4_f32(
            /*neg_a=*/false, a, /*neg_b=*/false, b,
            /*c_mod=*/(short)0, acc, /*reuse_a=*/false, /*reuse_b=*/false);
    }

    // Column N=0: lane 0 -> rows rowBase+0..7, lane 16 -> rows rowBase+8..15.
    if (m == 0) {
        const int base = rowBase + 8 * half;
#pragma unroll
        for (int r = 0; r < 8; ++r) {
            if (base + r < n_ent) ent_dot[base + r] = acc[r];
        }
    }
}

// ---------------------------------------------------------------------------
// Kernel 4: scores[n] = sigmoid(q_const + ent_dot[candidates[n]])
// Gather from the 40KB ent_dot table (L2-resident); pure bandwidth.
// ---------------------------------------------------------------------------
__global__ void score_kernel(const int* __restrict__ cand,
                             const float* __restrict__ ent_dot,
                             const float* __restrict__ qc,
                             float* __restrict__ out,
                             int n) {
    const int i = blockIdx.x * blockDim.x + threadIdx.x;
    if (i < n) {
        const float x = qc[0] + ent_dot[cand[i]];
        out[i] = 1.0f / (1.0f + expf(-x));
    }
}

// ---------------------------------------------------------------------------
extern "C" void kernel_launch(void* const* d_in, const int* in_sizes, int n_in,
                              void* d_out, int out_size, void* d_ws, size_t ws_size,
                              hipStream_t stream) {
    const int*   subject        = (const int*)d_in[0];
    const int*   relation       = (const int*)d_in[1];
    const int*   candidates     = (const int*)d_in[2];
    const float* entity_embed   = (const float*)d_in[3];
    const float* relation_embed = (const float*)d_in[4];
    const float* Wq             = (const float*)d_in[5];
    const float* bq             = (const float*)d_in[6];
    const float* Ws             = (const float*)d_in[7];
    const float* bs             = (const float*)d_in[8];
    float* out = (float*)d_out;

    const int n_cand = in_sizes[2];
    const int n_ent  = in_sizes[3] / DIM;   // 10000

    float* ws      = (float*)d_ws;
    float* u       = ws;                    // 2048 floats
    float* qc      = ws + 2048;             // 1 float
    float* ent_dot = ws + 2560;             // n_ent floats

    // 1) u = Wq @ wsq  (2048 rows, one wave each)
    wq_dot_kernel<<<(2 * DIM) / 8, 256, 0, stream>>>(Wq, Ws, u);

    // 2) q_const scalar
    qconst_kernel<<<1, 256, 0, stream>>>(entity_embed, relation_embed,
                                         subject, relation, u, Ws, bq, bs, qc);

    // 3) per-entity candidate-side dots via WMMA (16 rows / wave)
    const int n_blk = (n_ent + 15) / 16;
    ent_dot_kernel<<<n_blk, 32, 0, stream>>>(entity_embed, Ws + DIM, ent_dot, n_ent);

    // 4) gather + sigmoid over all candidates
    score_kernel<<<(n_cand + 255) / 256, 256, 0, stream>>>(candidates, ent_dot,
                                                           qc, out, n_cand);
}